// GNNLayer_89781996356232
// MI455X (gfx1250) — compile-verified
//
#include <hip/hip_runtime.h>
#include <math.h>

// Problem constants (from the reference)
constexpr int Bb = 4;
constexpr int Ll = 48;
constexpr int Kk = 32;
constexpr int Dd = 128;
constexpr int Nrows = Bb * Ll * Kk; // 6144 flattened (b,l,k) rows
constexpr float EPSf = 1e-5f;

constexpr int STRp = 129; // LDS row stride for 128-wide rows (129%64==1 -> conflict-free)
constexpr int STRl = 49;  // LDS row stride for 48-wide logit rows (odd -> conflict-free)

// CDNA5 WMMA fp32: D(16x16,f32) = A(16x4,f32) * B(4x16,f32) + C
typedef float v2f __attribute__((ext_vector_type(2)));
typedef float v8f __attribute__((ext_vector_type(8)));

__device__ __forceinline__ v8f wmma_f32_k4(v2f a, v2f b, v8f c) {
  // 8 args: (neg_a, A, neg_b, B, c_mod, C, reuse_a, reuse_b)
  return __builtin_amdgcn_wmma_f32_16x16x4_f32(false, a, false, b, (short)0, c,
                                               false, false);
}

__device__ __forceinline__ float wave_sum(float s) {
  #pragma unroll
  for (int off = 16; off > 0; off >>= 1) s += __shfl_xor(s, off, 32);
  return s;
}

// ---------------------------------------------------------------------------
// Kernel 0: out[c][r] = in[r][c]. One-time weight transpose so WMMA B
// fragments (two consecutive K-rows, same column) become contiguous 8 bytes.
// ---------------------------------------------------------------------------
__global__ __launch_bounds__(256) void transpose_kernel(
    const float* __restrict__ in, float* __restrict__ out, int R, int C) {
  int idx = blockIdx.x * 256 + threadIdx.x;
  if (idx < R * C) {
    int r = idx / C, c = idx % C;
    out[(size_t)c * R + r] = in[idx];
  }
}

// ---------------------------------------------------------------------------
// Kernel 1: per 16-row tile of h (rows = flattened b,l,k):
//   hidden1 = relu(h*Wm1 + bm1); messages = hidden1*Wm2 + bm2
//   hi_proj = h*Wa1[:D];  hj_proj = h*Wa1[D:]
// 256 threads = 8 waves; wave w owns output columns [16w,16w+16).
// Weights are pre-transposed: Wt[col][kin] -> B fragment = one b64 load.
// ---------------------------------------------------------------------------
__global__ __launch_bounds__(256) void proj_kernel(
    const float* __restrict__ h, const float* __restrict__ Wm1t,
    const float* __restrict__ bm1, const float* __restrict__ Wm2t,
    const float* __restrict__ bm2, const float* __restrict__ Wa1t,
    float* __restrict__ messages, float* __restrict__ hip_,
    float* __restrict__ hjp_) {
  __shared__ float Xs[16 * Dd]; // 8 KB: 16 rows of h
  __shared__ float H1[16 * Dd]; // 8 KB: relu(h*Wm1+bm1)

  const int tid = threadIdx.x;
  const int wave = tid >> 5;
  const int lane = tid & 31;
  const int m = lane & 15;        // row (A layout) == output column n (B/C layout)
  const int half = lane >> 4;     // lane group
  const int r0 = blockIdx.x * 16; // first global row of this tile
  const int col0 = wave * 16;

  // Cooperative coalesced load of the 16x128 h tile.
  #pragma unroll
  for (int t = 0; t < 8; ++t) {
    int f = tid + t * 256;
    Xs[f] = h[(size_t)r0 * Dd + f];
  }
  __syncthreads();

  // Transposed-weight row pointers for this wave's output column.
  const float* wm1c = Wm1t + (size_t)(col0 + m) * Dd;       // [col][kin]
  const float* wm2c = Wm2t + (size_t)(col0 + m) * Dd;       // [col][kin]
  const float* wa1c = Wa1t + (size_t)(col0 + m) * (2 * Dd); // [col][kin 0..255]

  // ---- GEMM phase 1: hidden1, hi_proj, hj_proj (shared A fragments) ----
  v8f acc_h1 = {0.f, 0.f, 0.f, 0.f, 0.f, 0.f, 0.f, 0.f};
  v8f acc_hi = acc_h1, acc_hj = acc_h1;
  #pragma unroll 4
  for (int kk = 0; kk < Dd; kk += 4) {
    const int kA = kk + 2 * half; // this lane's K columns (kA, kA+1)
    v2f a;
    a.x = Xs[m * Dd + kA];
    a.y = Xs[m * Dd + kA + 1];
    const v2f bm = *reinterpret_cast<const v2f*>(wm1c + kA);
    const v2f bi = *reinterpret_cast<const v2f*>(wa1c + kA);
    const v2f bj = *reinterpret_cast<const v2f*>(wa1c + Dd + kA);
    acc_h1 = wmma_f32_k4(a, bm, acc_h1);
    acc_hi = wmma_f32_k4(a, bi, acc_hi);
    acc_hj = wmma_f32_k4(a, bj, acc_hj);
  }

  // Epilogues: bias+relu -> LDS (hidden1); raw projections -> global.
  const float b1 = bm1[col0 + m];
  #pragma unroll
  for (int r = 0; r < 8; ++r) {
    const int row = r + 8 * half; // C/D layout: VGPR r -> M = r + 8*half
    H1[row * Dd + col0 + m] = fmaxf(acc_h1[r] + b1, 0.f);
    hip_[(size_t)(r0 + row) * Dd + col0 + m] = acc_hi[r];
    hjp_[(size_t)(r0 + row) * Dd + col0 + m] = acc_hj[r];
  }
  __syncthreads();

  // ---- GEMM phase 2: messages = hidden1 * Wm2 + bm2 ----
  v8f acc_m = {0.f, 0.f, 0.f, 0.f, 0.f, 0.f, 0.f, 0.f};
  #pragma unroll 4
  for (int kk = 0; kk < Dd; kk += 4) {
    const int kA = kk + 2 * half;
    v2f a;
    a.x = H1[m * Dd + kA];
    a.y = H1[m * Dd + kA + 1];
    const v2f b = *reinterpret_cast<const v2f*>(wm2c + kA);
    acc_m = wmma_f32_k4(a, b, acc_m);
  }
  const float b2 = bm2[col0 + m];
  #pragma unroll
  for (int r = 0; r < 8; ++r) {
    const int row = r + 8 * half;
    messages[(size_t)(r0 + row) * Dd + col0 + m] = acc_m[r] + b2;
  }
}

// ---------------------------------------------------------------------------
// Kernel 2: per (b,k): full 48x48 logit matrix + softmax.
//   logit[i,j] = sum_d relu((hp[i,d]+ba1[d]) + hj[j,d]) * Wa2[d]
// hj rows are reused by all 48 i's -> stage both projections in LDS once.
// ba1 folded into staged hp; +ba2 is softmax-invariant and dropped.
// No cross-lane reductions: each thread owns whole (i,j) dot products.
// Writes attn in (b,k,i,j) layout so kernel 3 reads it contiguously.
// ---------------------------------------------------------------------------
__global__ __launch_bounds__(256) void attn_kernel(
    const float* __restrict__ hip_, const float* __restrict__ hjp_,
    const float* __restrict__ ba1, const float* __restrict__ Wa2,
    float* __restrict__ attn2) {
  __shared__ float HP[Ll * STRp]; // hp + ba1, padded rows
  __shared__ float HJ[Ll * STRp]; // hj, padded rows
  __shared__ float WA[Dd];        // Wa2
  __shared__ float LG[Ll * STRl]; // logits -> softmax in place

  const int t = threadIdx.x;
  const int b = blockIdx.x / Kk;
  const int k = blockIdx.x % Kk;
  const int rowbase = b * (Ll * Kk) + k; // global row for index x: rowbase + x*Kk

  if (t < Dd) WA[t] = Wa2[t];
  #pragma unroll
  for (int s = 0; s < 6; ++s) {
    int f = t + s * 256;          // f in [0, 48*32): float4 slot
    int r = f >> 5, c = (f & 31) * 4;
    const size_t g = (size_t)(rowbase + r * Kk) * Dd + c;
    const float4 pv = *reinterpret_cast<const float4*>(hip_ + g);
    const float4 qv = *reinterpret_cast<const float4*>(hjp_ + g);
    const float4 bv = *reinterpret_cast<const float4*>(ba1 + c);
    HP[r * STRp + c + 0] = pv.x + bv.x;
    HP[r * STRp + c + 1] = pv.y + bv.y;
    HP[r * STRp + c + 2] = pv.z + bv.z;
    HP[r * STRp + c + 3] = pv.w + bv.w;
    HJ[r * STRp + c + 0] = qv.x;
    HJ[r * STRp + c + 1] = qv.y;
    HJ[r * STRp + c + 2] = qv.z;
    HJ[r * STRp + c + 3] = qv.w;
  }
  __syncthreads();

  // 2304 (i,j) dot products, 9 per thread; consecutive lanes -> consecutive j
  // (stride 129 floats == bank stride 1 -> conflict-free), hp row broadcast.
  #pragma unroll
  for (int s = 0; s < 9; ++s) {
    const int p = t + s * 256; // p in [0, 2304)
    const int i = p / Ll, j = p % Ll;
    const float* __restrict__ hpR = HP + i * STRp;
    const float* __restrict__ hjR = HJ + j * STRp;
    float acc = 0.f;
    #pragma unroll 8
    for (int d = 0; d < Dd; ++d) acc += fmaxf(hpR[d] + hjR[d], 0.f) * WA[d];
    LG[i * STRl + j] = acc;
  }
  __syncthreads();

  // Softmax over j, one thread per row i (stride-49 column walk: no conflicts).
  if (t < Ll) {
    float* __restrict__ row = LG + t * STRl;
    float mx = -1e30f;
    #pragma unroll 8
    for (int j = 0; j < Ll; ++j) mx = fmaxf(mx, row[j]);
    float sum = 0.f;
    #pragma unroll 8
    for (int j = 0; j < Ll; ++j) {
      const float e = __expf(row[j] - mx);
      row[j] = e;
      sum += e;
    }
    const float inv = 1.f / sum;
    #pragma unroll 8
    for (int j = 0; j < Ll; ++j) row[j] *= inv;
  }
  __syncthreads();

  // Coalesced write-out: attn2[(b*Kk+k)][i][j].
  float* __restrict__ outb = attn2 + (size_t)blockIdx.x * (Ll * Ll);
  for (int f = t; f < Ll * Ll; f += 256) outb[f] = LG[(f / Ll) * STRl + f % Ll];
}

// ---------------------------------------------------------------------------
// Kernel 3: per (b,k): aggregated(48x128) = attn(48x48) * messages(48x128)
// via WMMA, then fused residual + LayerNorm. 256 threads = 8 waves;
// wave w owns d-tile w, loops over the 3 i-tiles.
// ---------------------------------------------------------------------------
__global__ __launch_bounds__(256) void agg_ln_kernel(
    const float* __restrict__ h, const float* __restrict__ messages,
    const float* __restrict__ attn2, const float* __restrict__ gamma,
    const float* __restrict__ beta, float* __restrict__ out) {
  __shared__ float Ms[Ll * Dd]; // 24 KB messages[b, :, k, :]
  __shared__ float Ws[Ll * Ll]; //  9 KB attn2[(b,k), :, :]
  __shared__ float AG[Ll * Dd]; // 24 KB aggregated

  const int tid = threadIdx.x;
  const int wave = tid >> 5;
  const int lane = tid & 31;
  const int m = lane & 15;
  const int half = lane >> 4;
  const int b = blockIdx.x / Kk;
  const int k = blockIdx.x % Kk;

  // Stage messages rows for this (b,k): coalesced over d.
  #pragma unroll
  for (int t = 0; t < (Ll * Dd) / 256; ++t) {
    int f = tid + t * 256;
    int j = f / Dd, d = f % Dd;
    Ms[f] = messages[((size_t)(b * Ll + j) * Kk + k) * Dd + d];
  }
  // Stage attention weights: contiguous thanks to kernel 2's layout.
  const float* __restrict__ wsrc = attn2 + (size_t)blockIdx.x * (Ll * Ll);
  for (int f = tid; f < Ll * Ll; f += 256) Ws[f] = wsrc[f];
  __syncthreads();

  // WMMA aggregation: C(16x16) += A(16x4 of Ws) * B(4x16 of Ms), K-dim = j.
  const int d0 = wave * 16;
  for (int it = 0; it < 3; ++it) {
    const int i0 = it * 16;
    v8f acc = {0.f, 0.f, 0.f, 0.f, 0.f, 0.f, 0.f, 0.f};
    #pragma unroll
    for (int j0 = 0; j0 < Ll; j0 += 4) {
      const int jA = j0 + 2 * half;
      v2f a, bfrag;
      a.x = Ws[(i0 + m) * Ll + jA];
      a.y = Ws[(i0 + m) * Ll + jA + 1];
      bfrag.x = Ms[jA * Dd + d0 + m];
      bfrag.y = Ms[(jA + 1) * Dd + d0 + m];
      acc = wmma_f32_k4(a, bfrag, acc);
    }
    #pragma unroll
    for (int r = 0; r < 8; ++r) {
      AG[(i0 + r + 8 * half) * Dd + d0 + m] = acc[r];
    }
  }
  __syncthreads();

  // Fused residual + LayerNorm: wave w handles rows [6w, 6w+6).
  const int dL = lane * 4;
  const float4 gv = reinterpret_cast<const float4*>(gamma)[lane];
  const float4 bv = reinterpret_cast<const float4*>(beta)[lane];
  #pragma unroll
  for (int rr = 0; rr < 6; ++rr) {
    const int i = wave * 6 + rr;
    const size_t grow = ((size_t)(b * Ll + i) * Kk + k) * Dd;
    const float4 hv = *reinterpret_cast<const float4*>(h + grow + dL);
    float x0 = hv.x + AG[i * Dd + dL + 0];
    float x1 = hv.y + AG[i * Dd + dL + 1];
    float x2 = hv.z + AG[i * Dd + dL + 2];
    float x3 = hv.w + AG[i * Dd + dL + 3];
    const float mu = wave_sum(x0 + x1 + x2 + x3) * (1.f / Dd);
    x0 -= mu; x1 -= mu; x2 -= mu; x3 -= mu;
    const float var =
        wave_sum(x0 * x0 + x1 * x1 + x2 * x2 + x3 * x3) * (1.f / Dd);
    const float rstd = rsqrtf(var + EPSf);
    float4 o;
    o.x = x0 * rstd * gv.x + bv.x;
    o.y = x1 * rstd * gv.y + bv.y;
    o.z = x2 * rstd * gv.z + bv.z;
    o.w = x3 * rstd * gv.w + bv.w;
    *reinterpret_cast<float4*>(out + grow + dL) = o;
  }
}

// ---------------------------------------------------------------------------
extern "C" void kernel_launch(void* const* d_in, const int* in_sizes, int n_in,
                              void* d_out, int out_size, void* d_ws,
                              size_t ws_size, hipStream_t stream) {
  const float* h    = (const float*)d_in[0];
  const float* Wm1  = (const float*)d_in[1];
  const float* bm1  = (const float*)d_in[2];
  const float* Wm2  = (const float*)d_in[3];
  const float* bm2  = (const float*)d_in[4];
  const float* Wa1  = (const float*)d_in[5];
  const float* ba1  = (const float*)d_in[6];
  const float* Wa2  = (const float*)d_in[7];
  // d_in[8] = ba2: constant logit shift, softmax-invariant -> unused
  const float* gamma = (const float*)d_in[9];
  const float* beta  = (const float*)d_in[10];
  float* out = (float*)d_out;

  // Workspace layout (floats):
  // messages | hi_proj | hj_proj | attn2 | Wm1t | Wm2t | Wa1t
  float* ws = (float*)d_ws;
  float* messages = ws;
  float* hip_ = messages + (size_t)Nrows * Dd;
  float* hjp_ = hip_ + (size_t)Nrows * Dd;
  float* attn2 = hjp_ + (size_t)Nrows * Dd;       // Bb*Kk*Ll*Ll floats
  float* Wm1t = attn2 + (size_t)Bb * Kk * Ll * Ll;
  float* Wm2t = Wm1t + Dd * Dd;
  float* Wa1t = Wm2t + Dd * Dd;                   // 128 x 256

  transpose_kernel<<<(Dd * Dd + 255) / 256, 256, 0, stream>>>(Wm1, Wm1t, Dd, Dd);
  transpose_kernel<<<(Dd * Dd + 255) / 256, 256, 0, stream>>>(Wm2, Wm2t, Dd, Dd);
  transpose_kernel<<<(2 * Dd * Dd + 255) / 256, 256, 0, stream>>>(Wa1, Wa1t,
                                                                  2 * Dd, Dd);

  proj_kernel<<<Nrows / 16, 256, 0, stream>>>(h, Wm1t, bm1, Wm2t, bm2, Wa1t,
                                              messages, hip_, hjp_);
  attn_kernel<<<Bb * Kk, 256, 0, stream>>>(hip_, hjp_, ba1, Wa2, attn2);
  agg_ln_kernel<<<Bb * Kk, 256, 0, stream>>>(h, messages, attn2, gamma, beta,
                                             out);
}